// BatchNormSPD_5781025980864
// MI455X (gfx1250) — compile-verified
//
#include <hip/hip_runtime.h>
#include <hip/hip_bf16.h>
#include <math.h>

// ---------------------------------------------------------------------------
// SPD BatchNorm (torchspdnet BaryGeom + congruence) for MI455X / gfx1250.
// All matrix functions (sqrtm, invsqrtm, logm, expm) are implemented as
// matmul-only iterations so the hot path is V_WMMA_F32_16X16X4_F32.
// One wave32 owns one 32x32 matrix; matrices live in padded LDS (stride 33).
// ---------------------------------------------------------------------------

typedef float v2f __attribute__((ext_vector_type(2)));
typedef float v8f __attribute__((ext_vector_type(8)));

#define LSTRIDE 33
#define MSIZE   (32 * LSTRIDE)

#define NS_ITERS        12   // Newton-Schulz iterations for sqrt/invsqrt
#define LOG_SQRT_ROUNDS 4    // repeated square roots before Mercator
#define MERC_TERMS      8    // Mercator series terms for log(I+E)
#define EXP_TAYLOR      8    // Taylor terms for expm
#define EXP_SQUARINGS   6    // expm scaling: t / 2^6, then 6 squarings

// ---------------------------------------------------------------------------
// One-wave 32x32 fp32 matmul: C = A @ B using WMMA 16x16x4 tiles.
// A/B/C are LDS matrices with row stride LSTRIDE.
// Fragment layouts per CDNA5 ISA 7.12.2:
//   A 16x4 f32 : lanes 0-15 -> M=lane, v.x=K0 v.y=K1 ; lanes 16-31 -> K2,K3
//   B 4x16 f32 : row striped across lanes within a VGPR (mirrored)
//   C 16x16 f32: VGPR r -> M=r (lanes 0-15), M=r+8 (lanes 16-31), N=lane%16
// ---------------------------------------------------------------------------
__device__ __forceinline__ void mm32(const float* __restrict__ A,
                                     const float* __restrict__ B,
                                     float* __restrict__ C, int lane) {
  const int half = lane >> 4;   // 0 or 1
  const int l16  = lane & 15;
  v8f acc00 = {}, acc01 = {}, acc10 = {}, acc11 = {};
#pragma unroll
  for (int kb = 0; kb < 32; kb += 4) {
    v2f a0, a1, b0, b1;
    const int ka = kb + 2 * half;
    a0.x = A[(l16)*LSTRIDE + ka];
    a0.y = A[(l16)*LSTRIDE + ka + 1];
    a1.x = A[(l16 + 16) * LSTRIDE + ka];
    a1.y = A[(l16 + 16) * LSTRIDE + ka + 1];
    b0.x = B[(ka)*LSTRIDE + l16];
    b0.y = B[(ka + 1) * LSTRIDE + l16];
    b1.x = B[(ka)*LSTRIDE + l16 + 16];
    b1.y = B[(ka + 1) * LSTRIDE + l16 + 16];
    acc00 = __builtin_amdgcn_wmma_f32_16x16x4_f32(false, a0, false, b0,
                                                  (short)0, acc00, false, false);
    acc01 = __builtin_amdgcn_wmma_f32_16x16x4_f32(false, a0, false, b1,
                                                  (short)0, acc01, false, false);
    acc10 = __builtin_amdgcn_wmma_f32_16x16x4_f32(false, a1, false, b0,
                                                  (short)0, acc10, false, false);
    acc11 = __builtin_amdgcn_wmma_f32_16x16x4_f32(false, a1, false, b1,
                                                  (short)0, acc11, false, false);
  }
#pragma unroll
  for (int r = 0; r < 8; ++r) {
    C[(r + 8 * half) * LSTRIDE + l16]            = acc00[r];
    C[(r + 8 * half) * LSTRIDE + l16 + 16]       = acc01[r];
    C[(r + 8 * half + 16) * LSTRIDE + l16]       = acc10[r];
    C[(r + 8 * half + 16) * LSTRIDE + l16 + 16]  = acc11[r];
  }
}

// Elementwise helpers: lane owns column `lane`, rows k = 0..31.
__device__ __forceinline__ void wset_ident(float* M, int lane, float d) {
  for (int k = 0; k < 32; ++k) M[k * LSTRIDE + lane] = (k == lane) ? d : 0.0f;
}
__device__ __forceinline__ void wcopy(float* D, const float* S, int lane) {
  for (int k = 0; k < 32; ++k) D[k * LSTRIDE + lane] = S[k * LSTRIDE + lane];
}
__device__ __forceinline__ void wscale(float* D, const float* S, int lane, float a) {
  for (int k = 0; k < 32; ++k) D[k * LSTRIDE + lane] = a * S[k * LSTRIDE + lane];
}
__device__ __forceinline__ void waxpy(float* D, const float* S, int lane, float a) {
  for (int k = 0; k < 32; ++k) D[k * LSTRIDE + lane] += a * S[k * LSTRIDE + lane];
}
__device__ __forceinline__ void wthree_minus(float* T, int lane) {  // T = 3I - T
  for (int k = 0; k < 32; ++k) {
    float v = T[k * LSTRIDE + lane];
    T[k * LSTRIDE + lane] = ((k == lane) ? 3.0f : 0.0f) - v;
  }
}
__device__ __forceinline__ void wsym(float* M, int lane) {  // M = (M + M^T)/2
  float va[32];
#pragma unroll
  for (int k = 0; k < 32; ++k)
    va[k] = 0.5f * (M[k * LSTRIDE + lane] + M[lane * LSTRIDE + k]);
#pragma unroll
  for (int k = 0; k < 32; ++k) M[k * LSTRIDE + lane] = va[k];
}
__device__ __forceinline__ float wfrob(const float* M, int lane) {
  float s = 0.0f;
  for (int k = 0; k < 32; ++k) {
    float v = M[k * LSTRIDE + lane];
    s += v * v;
  }
#pragma unroll
  for (int m = 16; m >= 1; m >>= 1) s += __shfl_xor(s, m, 32);
  return sqrtf(s);
}

// Coupled Newton-Schulz: Y0 (eigs in (0,2)) -> Y=Y0^{1/2}, Z=Y0^{-1/2}.
__device__ __forceinline__ void ns_sqrt(float* Y, float* Z, float* T, float* W,
                                        int lane, int iters) {
  wset_ident(Z, lane, 1.0f);
  for (int it = 0; it < iters; ++it) {
    mm32(Z, Y, T, lane);     // T = Z@Y
    wthree_minus(T, lane);   // T = 3I - Z@Y
    mm32(Y, T, W, lane);
    wscale(Y, W, lane, 0.5f);  // Y = Y@T / 2
    mm32(T, Z, W, lane);
    wscale(Z, W, lane, 0.5f);  // Z = T@Z / 2
  }
}

// Arbitrary SPD in Y: Y <- sqrtm, Z <- invsqrtm (Frobenius normalization).
__device__ __forceinline__ void spd_sqrt_invsqrt(float* Y, float* Z, float* T,
                                                 float* W, int lane) {
  float s = wfrob(Y, lane);
  wscale(Y, Y, lane, 1.0f / s);
  ns_sqrt(Y, Z, T, W, lane, NS_ITERS);
  float sq = sqrtf(s);
  wscale(Y, Y, lane, sq);
  wscale(Z, Z, lane, 1.0f / sq);
}

// logm of SPD matrix in Y -> result in L.  logm(S) = 2^k log(A^{1/2^k}) + log(|S|_F) I
__device__ __forceinline__ void spd_logm(float* Y, float* Z, float* T, float* W,
                                         float* L, int lane) {
  float s0 = wfrob(Y, lane);
  wscale(Y, Y, lane, 1.0f / s0);  // eigenvalues in (0,1] -> NS converges
  for (int r = 0; r < LOG_SQRT_ROUNDS; ++r)
    ns_sqrt(Y, Z, T, W, lane, NS_ITERS);  // Y <- Y^{1/2}, eigs stay in (0,1]
  // E = Y - I (diagonal owned by lane==k)
  Y[lane * LSTRIDE + lane] -= 1.0f;
  // Mercator: L = E - E^2/2 + E^3/3 - ...
  wcopy(L, Y, lane);
  wcopy(Z, Y, lane);  // Z = current power E^m
  float sign = -1.0f;
  for (int m = 2; m <= MERC_TERMS; ++m) {
    mm32(Z, Y, W, lane);
    wcopy(Z, W, lane);
    waxpy(L, W, lane, sign / (float)m);
    sign = -sign;
  }
  const float sc = (float)(1 << LOG_SQRT_ROUNDS);
  const float lg = logf(s0);
  for (int k = 0; k < 32; ++k) {
    float v = L[k * LSTRIDE + lane] * sc;
    if (k == lane) v += lg;
    L[k * LSTRIDE + lane] = v;
  }
}

// ------------------------------- kernels -----------------------------------
__global__ void k_zero(float* p, int n) {
  for (int i = threadIdx.x + blockIdx.x * blockDim.x; i < n;
       i += blockDim.x * gridDim.x)
    p[i] = 0.0f;
}

// Sum of X over the batch -> gsum (1024 floats).
__global__ void k_mean(const float* __restrict__ X, float* __restrict__ gsum,
                       int Nb) {
  const int t = threadIdx.x;  // 256 threads
  float acc[4] = {0.f, 0.f, 0.f, 0.f};
  for (int m = blockIdx.x; m < Nb; m += gridDim.x) {
    const float* Xm = X + (size_t)m * 1024;
#pragma unroll
    for (int j = 0; j < 4; ++j) acc[j] += Xm[t + 256 * j];
  }
#pragma unroll
  for (int j = 0; j < 4; ++j) atomicAdd(&gsum[t + 256 * j], acc[j]);
}

// g = gsum/Nb ; gs = sqrtm(g) ; gis = invsqrtm(g)
__global__ __launch_bounds__(32) void k_mean_sqrt(const float* __restrict__ gsum,
                                                  float* __restrict__ gs_out,
                                                  float* __restrict__ gis_out,
                                                  float invNb) {
  __shared__ float Y[MSIZE], Z[MSIZE], T[MSIZE], W[MSIZE];
  const int lane = threadIdx.x;
  for (int k = 0; k < 32; ++k)
    Y[k * LSTRIDE + lane] = gsum[k * 32 + lane] * invNb;
  spd_sqrt_invsqrt(Y, Z, T, W, lane);
  for (int k = 0; k < 32; ++k) {
    gs_out[k * 32 + lane]  = Y[k * LSTRIDE + lane];
    gis_out[k * 32 + lane] = Z[k * LSTRIDE + lane];
  }
}

// tsum += sum_i logm(sym(gis @ X_i @ gis))
__global__ __launch_bounds__(32) void k_logmean(const float* __restrict__ X,
                                                const float* __restrict__ gis,
                                                float* __restrict__ tsum, int Nb) {
  __shared__ float B0[MSIZE], B1[MSIZE], B2[MSIZE], B3[MSIZE], B4[MSIZE];
  const int lane = threadIdx.x;
  float acc[32];
#pragma unroll
  for (int k = 0; k < 32; ++k) acc[k] = 0.0f;

  for (int m = blockIdx.x; m < Nb; m += gridDim.x) {
    const float* Xm = X + (size_t)m * 1024;
    for (int k = 0; k < 32; ++k) B2[k * LSTRIDE + lane] = Xm[k * 32 + lane];
    for (int k = 0; k < 32; ++k) B1[k * LSTRIDE + lane] = gis[k * 32 + lane];
    mm32(B1, B2, B3, lane);  // gis @ X
    mm32(B3, B1, B0, lane);  // (gis @ X) @ gis
    wsym(B0, lane);
    spd_logm(B0, B1, B2, B3, B4, lane);  // log-map in B4
#pragma unroll
    for (int k = 0; k < 32; ++k) acc[k] += B4[k * LSTRIDE + lane];
  }
#pragma unroll
  for (int k = 0; k < 32; ++k) atomicAdd(&tsum[k * 32 + lane], acc[k]);
}

// g2 = sym(gs @ expm(tsum/Nb) @ gs); gis2 = invsqrtm(g2); wsm = sqrtm(weight)
__global__ __launch_bounds__(32) void k_update(const float* __restrict__ tsum,
                                               const float* __restrict__ gs,
                                               const float* __restrict__ weight,
                                               float* __restrict__ gis2_out,
                                               float* __restrict__ wsm_out,
                                               float invNb) {
  __shared__ float B0[MSIZE], B1[MSIZE], B2[MSIZE], B3[MSIZE], B4[MSIZE];
  const int lane = threadIdx.x;
  // S = (t/Nb) / 2^EXP_SQUARINGS
  const float sc = invNb * (1.0f / (float)(1 << EXP_SQUARINGS));
  for (int k = 0; k < 32; ++k)
    B0[k * LSTRIDE + lane] = tsum[k * 32 + lane] * sc;
  // Taylor: E = I + S + S^2/2! + ...   (E in B3, powers in B1, tmp B2)
  wset_ident(B3, lane, 1.0f);
  waxpy(B3, B0, lane, 1.0f);
  wcopy(B1, B0, lane);
  for (int t = 2; t <= EXP_TAYLOR; ++t) {
    mm32(B1, B0, B2, lane);
    wscale(B1, B2, lane, 1.0f / (float)t);
    waxpy(B3, B1, lane, 1.0f);
  }
  for (int q = 0; q < EXP_SQUARINGS; ++q) {  // E <- E@E
    mm32(B3, B3, B2, lane);
    wcopy(B3, B2, lane);
  }
  // g2 = sym(gs @ E @ gs)
  for (int k = 0; k < 32; ++k) B1[k * LSTRIDE + lane] = gs[k * 32 + lane];
  mm32(B1, B3, B2, lane);
  mm32(B2, B1, B0, lane);
  wsym(B0, lane);
  // gis2 = invsqrtm(g2)
  spd_sqrt_invsqrt(B0, B1, B2, B3, lane);
  for (int k = 0; k < 32; ++k) gis2_out[k * 32 + lane] = B1[k * LSTRIDE + lane];
  // wsm = sqrtm(weight)
  for (int k = 0; k < 32; ++k) B0[k * LSTRIDE + lane] = weight[k * 32 + lane];
  spd_sqrt_invsqrt(B0, B1, B2, B3, lane);
  for (int k = 0; k < 32; ++k) wsm_out[k * 32 + lane] = B0[k * LSTRIDE + lane];
}

// Xn_i = wsm @ (gis2 @ X_i @ gis2) @ wsm
__global__ __launch_bounds__(32) void k_transform(const float* __restrict__ X,
                                                  const float* __restrict__ gis2,
                                                  const float* __restrict__ wsm,
                                                  float* __restrict__ out, int Nb) {
  __shared__ float G[MSIZE], Wm[MSIZE], B2[MSIZE], B3[MSIZE], B4[MSIZE];
  const int lane = threadIdx.x;
  for (int k = 0; k < 32; ++k) G[k * LSTRIDE + lane] = gis2[k * 32 + lane];
  for (int k = 0; k < 32; ++k) Wm[k * LSTRIDE + lane] = wsm[k * 32 + lane];
  for (int m = blockIdx.x; m < Nb; m += gridDim.x) {
    const float* Xm = X + (size_t)m * 1024;
    float* Om = out + (size_t)m * 1024;
    for (int k = 0; k < 32; ++k) B2[k * LSTRIDE + lane] = Xm[k * 32 + lane];
    mm32(G, B2, B3, lane);    // gis2 @ X
    mm32(B3, G, B4, lane);    // Xc
    mm32(Wm, B4, B3, lane);   // wsm @ Xc
    mm32(B3, Wm, B2, lane);   // Xn
    for (int k = 0; k < 32; ++k) Om[k * 32 + lane] = B2[k * LSTRIDE + lane];
  }
}

// ------------------------------- launcher ----------------------------------
extern "C" void kernel_launch(void* const* d_in, const int* in_sizes, int n_in,
                              void* d_out, int out_size, void* d_ws, size_t ws_size,
                              hipStream_t stream) {
  const float* X      = (const float*)d_in[0];
  const float* weight = (const float*)d_in[1];
  float* out = (float*)d_out;
  float* ws  = (float*)d_ws;

  const int Nb = in_sizes[0] / 1024;  // N*h matrices of 32x32
  if (Nb <= 0) return;
  const float invNb = 1.0f / (float)Nb;

  float* gsum = ws;           // 1024
  float* tsum = ws + 1024;    // 1024
  float* gs   = ws + 2048;    // 1024
  float* gis  = ws + 3072;    // 1024
  float* gis2 = ws + 4096;    // 1024
  float* wsm  = ws + 5120;    // 1024

  k_zero<<<4, 256, 0, stream>>>(ws, 2048);
  k_mean<<<256, 256, 0, stream>>>(X, gsum, Nb);
  k_mean_sqrt<<<1, 32, 0, stream>>>(gsum, gs, gis, invNb);
  int g3 = Nb < 512 ? Nb : 512;
  k_logmean<<<g3, 32, 0, stream>>>(X, gis, tsum, Nb);
  k_update<<<1, 32, 0, stream>>>(tsum, gs, weight, gis2, wsm, invNb);
  int g5 = Nb < 2048 ? Nb : 2048;
  k_transform<<<g5, 32, 0, stream>>>(X, gis2, wsm, out, Nb);
}